// Convolution_12386685681676
// MI455X (gfx1250) — compile-verified
//
#include <hip/hip_runtime.h>

typedef float v2f __attribute__((ext_vector_type(2)));
typedef float v8f __attribute__((ext_vector_type(8)));

#define N_NODES 20000
#define N_EDGES 640000

__device__ __forceinline__ v8f zero8() {
  v8f v = {0.f, 0.f, 0.f, 0.f, 0.f, 0.f, 0.f, 0.f};
  return v;
}

// D = A(16x4 f32) * B(4x16 f32) + C(16x16 f32)   -> v_wmma_f32_16x16x4_f32
__device__ __forceinline__ v8f wmma4(v2f a, v2f b, v8f c) {
  return __builtin_amdgcn_wmma_f32_16x16x4_f32(false, a, false, b, (short)0, c,
                                               false, false);
}

__device__ __forceinline__ float gelu_tanh(float x) {
  float x3 = x * x * x;
  return 0.5f * x * (1.0f + tanhf(0.7978845608028654f * (x + 0.044715f * x3)));
}

// ---------------------------------------------------------------------------
// Kernel A: per-node equivariant linear.
//   fs = (node_s @ Wsf) * inv_mul     ss = (node_s @ Wss) * inv_mul
//   fv[n][c][v] = sum_u node_v[n][u][c] * Wvf[u][v] * inv_mul   (ditto sv)
// One wave per 16-node tile. fv/sv stored [N,3,32] component-major.
// ---------------------------------------------------------------------------
__global__ __launch_bounds__(128) void node_linear_kernel(
    const float* __restrict__ node_s, const float* __restrict__ node_v,
    const float* __restrict__ Wsf, const float* __restrict__ Wss,
    const float* __restrict__ Wvf, const float* __restrict__ Wvs,
    float* __restrict__ fs, float* __restrict__ fvb, float* __restrict__ ss,
    float* __restrict__ svb) {
  const int lane = threadIdx.x & 31;
  const int wave = threadIdx.x >> 5;
  const int tile = blockIdx.x * 4 + wave;
  if (tile >= N_NODES / 16) return;
  const int n0 = tile * 16;
  const int l15 = lane & 15, lh = lane >> 4;
  const float inv_mul = 0.17677669529663687f;  // 1/sqrt(32)

  // ---- scalar irrep: [16,32] x [32,32] for feat and self
  {
    v8f aF0 = zero8(), aF1 = zero8(), aS0 = zero8(), aS1 = zero8();
    const float* arow = node_s + (size_t)(n0 + l15) * 32;
#pragma unroll
    for (int ks = 0; ks < 8; ++ks) {
      int k = ks * 4 + lh * 2;
      v2f a = *reinterpret_cast<const v2f*>(arow + k);
      v2f bf0, bf1, bs0, bs1;
      bf0.x = Wsf[k * 32 + l15];      bf0.y = Wsf[(k + 1) * 32 + l15];
      bf1.x = Wsf[k * 32 + 16 + l15]; bf1.y = Wsf[(k + 1) * 32 + 16 + l15];
      bs0.x = Wss[k * 32 + l15];      bs0.y = Wss[(k + 1) * 32 + l15];
      bs1.x = Wss[k * 32 + 16 + l15]; bs1.y = Wss[(k + 1) * 32 + 16 + l15];
      aF0 = wmma4(a, bf0, aF0);
      aF1 = wmma4(a, bf1, aF1);
      aS0 = wmma4(a, bs0, aS0);
      aS1 = wmma4(a, bs1, aS1);
    }
#pragma unroll
    for (int r = 0; r < 8; ++r) {
      int node = n0 + r + 8 * lh;
      fs[(size_t)node * 32 + l15]      = aF0[r] * inv_mul;
      fs[(size_t)node * 32 + 16 + l15] = aF1[r] * inv_mul;
      ss[(size_t)node * 32 + l15]      = aS0[r] * inv_mul;
      ss[(size_t)node * 32 + 16 + l15] = aS1[r] * inv_mul;
    }
  }

  // ---- vector irrep: one GEMM per Cartesian component
#pragma unroll
  for (int c = 0; c < 3; ++c) {
    v8f aF0 = zero8(), aF1 = zero8(), aS0 = zero8(), aS1 = zero8();
    const float* vrow = node_v + (size_t)(n0 + l15) * 96 + c;  // [N,32,3]
#pragma unroll
    for (int ks = 0; ks < 8; ++ks) {
      int k = ks * 4 + lh * 2;
      v2f a;
      a.x = vrow[k * 3];
      a.y = vrow[(k + 1) * 3];
      v2f bf0, bf1, bs0, bs1;
      bf0.x = Wvf[k * 32 + l15];      bf0.y = Wvf[(k + 1) * 32 + l15];
      bf1.x = Wvf[k * 32 + 16 + l15]; bf1.y = Wvf[(k + 1) * 32 + 16 + l15];
      bs0.x = Wvs[k * 32 + l15];      bs0.y = Wvs[(k + 1) * 32 + l15];
      bs1.x = Wvs[k * 32 + 16 + l15]; bs1.y = Wvs[(k + 1) * 32 + 16 + l15];
      aF0 = wmma4(a, bf0, aF0);
      aF1 = wmma4(a, bf1, aF1);
      aS0 = wmma4(a, bs0, aS0);
      aS1 = wmma4(a, bs1, aS1);
    }
#pragma unroll
    for (int r = 0; r < 8; ++r) {
      int node = n0 + r + 8 * lh;
      fvb[((size_t)node * 3 + c) * 32 + l15]      = aF0[r] * inv_mul;
      fvb[((size_t)node * 3 + c) * 32 + 16 + l15] = aF1[r] * inv_mul;
      svb[((size_t)node * 3 + c) * 32 + l15]      = aS0[r] * inv_mul;
      svb[((size_t)node * 3 + c) * 32 + 16 + l15] = aS1[r] * inv_mul;
    }
  }
}

// ---------------------------------------------------------------------------
// Kernel B: per-edge radial MLP (3 GEMMs on WMMA) + uvu tensor product +
// atomic scatter. One wave per 16-edge tile, 2 waves (64 threads) per block.
// Weights live in LDS pre-permuted into WMMA B-fragment order; MLP
// normalization constants are folded into the staged weights.
// ---------------------------------------------------------------------------
__global__ __launch_bounds__(64) void edge_kernel(
    const float* __restrict__ edge_s, const float* __restrict__ edge_v,
    const float* __restrict__ edge_rad, const int* __restrict__ esrc,
    const int* __restrict__ edst, const float* __restrict__ Wmlp0,
    const float* __restrict__ Wmlp1, const float* __restrict__ Wi0,
    const float* __restrict__ Wi1, const float* __restrict__ Wi2,
    const float* __restrict__ Wi3, const float* __restrict__ fs,
    const float* __restrict__ fvb, float* __restrict__ ns,
    float* __restrict__ nv) {
  // LDS: [0,1024)   Wmlp0 fragments  (KS=4,  NT=4)
  //      [1024,5120) Wmlp1 fragments (KS=16, NT=4)
  //      [5120,13312) Wcat fragments (KS=16, NT=8)  cols: w0|w1|w2|w3
  //      [13312, +2*1056) per-wave h transpose buffers (16 rows, stride 66)
  __shared__ float lds[13312 + 2 * 1056];
  const int tid = threadIdx.x;

  for (int idx = tid; idx < 1024; idx += 64) {
    int j = idx & 1, l = (idx >> 1) & 31, f = idx >> 6;
    int k = (f >> 2) * 4 + (l >> 4) * 2 + j;
    int n = (f & 3) * 16 + (l & 15);
    lds[idx] = Wmlp0[k * 64 + n] * 0.25f;  // 1/sqrt(16)
  }
  for (int idx = tid; idx < 4096; idx += 64) {
    int j = idx & 1, l = (idx >> 1) & 31, f = idx >> 6;
    int k = (f >> 2) * 4 + (l >> 4) * 2 + j;
    int n = (f & 3) * 16 + (l & 15);
    lds[1024 + idx] = Wmlp1[k * 64 + n] * 0.125f;  // 1/sqrt(64)
  }
  for (int idx = tid; idx < 8192; idx += 64) {
    int j = idx & 1, l = (idx >> 1) & 31, f = idx >> 6;
    int k = (f >> 3) * 4 + (l >> 4) * 2 + j;
    int n = (f & 7) * 16 + (l & 15);
    const float* Wi = (n < 32) ? Wi0 : (n < 64) ? Wi1 : (n < 96) ? Wi2 : Wi3;
    lds[5120 + idx] = Wi[k * 32 + (n & 31)] * 0.125f;  // 1/sqrt(64)
  }
  __syncthreads();

  const int lane = tid & 31, wave = tid >> 5;
  const int l15 = lane & 15, lh = lane >> 4;
  const int tile = blockIdx.x * 2 + wave;
  const int e0 = tile * 16;
  float* hbuf = lds + 13312 + wave * 1056;

  // ---- GEMM1: h1 = gelu(X[16,16] @ Wmlp0[16,64] / 4)
  v8f h1[4];
#pragma unroll
  for (int nt = 0; nt < 4; ++nt) h1[nt] = zero8();
  const float* Xrow = edge_rad + ((size_t)e0 + l15) * 16;
#pragma unroll
  for (int ks = 0; ks < 4; ++ks) {
    int k = ks * 4 + lh * 2;
    v2f a = *reinterpret_cast<const v2f*>(Xrow + k);
#pragma unroll
    for (int nt = 0; nt < 4; ++nt) {
      v2f b = *reinterpret_cast<const v2f*>(&lds[(ks * 4 + nt) * 64 + lane * 2]);
      h1[nt] = wmma4(a, b, h1[nt]);
    }
  }
#pragma unroll
  for (int nt = 0; nt < 4; ++nt)
#pragma unroll
    for (int r = 0; r < 8; ++r)
      hbuf[(r + 8 * lh) * 66 + nt * 16 + l15] = gelu_tanh(h1[nt][r]);

  // ---- GEMM2: h2 = gelu(h1 @ Wmlp1 / 8)
  v8f h2[4];
#pragma unroll
  for (int nt = 0; nt < 4; ++nt) h2[nt] = zero8();
#pragma unroll
  for (int ks = 0; ks < 16; ++ks) {
    int k = ks * 4 + lh * 2;
    v2f a = *reinterpret_cast<const v2f*>(&hbuf[l15 * 66 + k]);
#pragma unroll
    for (int nt = 0; nt < 4; ++nt) {
      v2f b =
          *reinterpret_cast<const v2f*>(&lds[1024 + (ks * 4 + nt) * 64 + lane * 2]);
      h2[nt] = wmma4(a, b, h2[nt]);
    }
  }
#pragma unroll
  for (int nt = 0; nt < 4; ++nt)
#pragma unroll
    for (int r = 0; r < 8; ++r)
      hbuf[(r + 8 * lh) * 66 + nt * 16 + l15] = gelu_tanh(h2[nt][r]);

  // ---- GEMM3: [w0|w1|w2|w3] = h2 @ Wcat[64,128] / 8
  v8f w[8];
#pragma unroll
  for (int nt = 0; nt < 8; ++nt) w[nt] = zero8();
#pragma unroll
  for (int ks = 0; ks < 16; ++ks) {
    int k = ks * 4 + lh * 2;
    v2f a = *reinterpret_cast<const v2f*>(&hbuf[l15 * 66 + k]);
#pragma unroll
    for (int nt = 0; nt < 8; ++nt) {
      v2f b =
          *reinterpret_cast<const v2f*>(&lds[5120 + (ks * 8 + nt) * 64 + lane * 2]);
      w[nt] = wmma4(a, b, w[nt]);
    }
  }

  // ---- uvu tensor product + atomic scatter. C-layout: vgpr r, lane L holds
  //      (edge m = r + 8*(L/16), channel u = L%16 [+16 for odd ntile]).
  const float inv_s3 = 0.5773502691896258f;  // 1/sqrt(3)
#pragma unroll
  for (int r = 0; r < 8; ++r) {
    int m = r + 8 * lh;
    int e = e0 + m;
    float se = edge_s[e];
    float vx = edge_v[(size_t)e * 3 + 0];
    float vy = edge_v[(size_t)e * 3 + 1];
    float vz = edge_v[(size_t)e * 3 + 2];
    int src = esrc[e];
    int dst = edst[e];
    const float* fr = fs + (size_t)src * 32;
    const float* vr = fvb + (size_t)src * 96;  // [3][32]
    float esA = fr[l15], esB = fr[16 + l15];
    float exA = vr[l15], exB = vr[16 + l15];
    float eyA = vr[32 + l15], eyB = vr[48 + l15];
    float ezA = vr[64 + l15], ezB = vr[80 + l15];
    float w0A = w[0][r], w0B = w[1][r];
    float w1A = w[2][r], w1B = w[3][r];
    float w2A = w[4][r], w2B = w[5][r];
    float w3A = w[6][r], w3B = w[7][r];
    float dA = exA * vx + eyA * vy + ezA * vz;
    float dB = exB * vx + eyB * vy + ezB * vz;

    float* nsr = ns + (size_t)dst * 64;
    atomicAdd(nsr + l15, w0A * esA * se);            // m0, ch 0..31
    atomicAdd(nsr + 16 + l15, w0B * esB * se);
    atomicAdd(nsr + 32 + l15, w3A * dA * inv_s3);    // m3, ch 32..63
    atomicAdd(nsr + 48 + l15, w3B * dB * inv_s3);

    float t1A = w1A * esA, t1B = w1B * esB;
    float vcs[3] = {vx, vy, vz};
    float evA[3] = {exA, eyA, ezA};
    float evB[3] = {exB, eyB, ezB};
    float* nvr = nv + (size_t)dst * 192;  // [3][64]
#pragma unroll
    for (int c = 0; c < 3; ++c) {
      atomicAdd(nvr + c * 64 + l15, t1A * vcs[c]);         // m1, ch 0..31
      atomicAdd(nvr + c * 64 + 16 + l15, t1B * vcs[c]);
      atomicAdd(nvr + c * 64 + 32 + l15, w2A * evA[c] * se);  // m2, ch 32..63
      atomicAdd(nvr + c * 64 + 48 + l15, w2B * evB[c] * se);
    }
  }
}

// ---------------------------------------------------------------------------
// Kernel C: second equivariant linear + self-connection blend.
// out_s = cos*ss + sin*inv_nb*inv_mid*(ns @ W2s)
// out_v = cos*sv + sin*inv_nb*inv_mid*(nv @ W2v)   (per component)
// ---------------------------------------------------------------------------
__global__ __launch_bounds__(128) void out_linear_kernel(
    const float* __restrict__ ns, const float* __restrict__ nv,
    const float* __restrict__ W2s, const float* __restrict__ W2v,
    const float* __restrict__ ss, const float* __restrict__ svb,
    float* __restrict__ out) {
  const int lane = threadIdx.x & 31;
  const int wave = threadIdx.x >> 5;
  const int tile = blockIdx.x * 4 + wave;
  if (tile >= N_NODES / 16) return;
  const int n0 = tile * 16;
  const int l15 = lane & 15, lh = lane >> 4;
  const float cosA = 0.9238795325112867f;                      // cos(pi/8)
  const float scale = 0.3826834323650898f * 0.125f * 0.17677669529663687f;
  float* outs = out;
  float* outv = out + (size_t)N_NODES * 32;

  // ---- scalars: [16,64] x [64,32]
  {
    v8f a0 = zero8(), a1 = zero8();
    const float* arow = ns + (size_t)(n0 + l15) * 64;
#pragma unroll
    for (int ks = 0; ks < 16; ++ks) {
      int k = ks * 4 + lh * 2;
      v2f a = *reinterpret_cast<const v2f*>(arow + k);
      v2f b0, b1;
      b0.x = W2s[k * 32 + l15];      b0.y = W2s[(k + 1) * 32 + l15];
      b1.x = W2s[k * 32 + 16 + l15]; b1.y = W2s[(k + 1) * 32 + 16 + l15];
      a0 = wmma4(a, b0, a0);
      a1 = wmma4(a, b1, a1);
    }
#pragma unroll
    for (int r = 0; r < 8; ++r) {
      int node = n0 + r + 8 * lh;
      outs[(size_t)node * 32 + l15] =
          cosA * ss[(size_t)node * 32 + l15] + scale * a0[r];
      outs[(size_t)node * 32 + 16 + l15] =
          cosA * ss[(size_t)node * 32 + 16 + l15] + scale * a1[r];
    }
  }

  // ---- vectors per component
#pragma unroll
  for (int c = 0; c < 3; ++c) {
    v8f a0 = zero8(), a1 = zero8();
    const float* arow = nv + ((size_t)(n0 + l15) * 3 + c) * 64;
#pragma unroll
    for (int ks = 0; ks < 16; ++ks) {
      int k = ks * 4 + lh * 2;
      v2f a = *reinterpret_cast<const v2f*>(arow + k);
      v2f b0, b1;
      b0.x = W2v[k * 32 + l15];      b0.y = W2v[(k + 1) * 32 + l15];
      b1.x = W2v[k * 32 + 16 + l15]; b1.y = W2v[(k + 1) * 32 + 16 + l15];
      a0 = wmma4(a, b0, a0);
      a1 = wmma4(a, b1, a1);
    }
#pragma unroll
    for (int r = 0; r < 8; ++r) {
      int node = n0 + r + 8 * lh;
      outv[((size_t)node * 32 + l15) * 3 + c] =
          cosA * svb[((size_t)node * 3 + c) * 32 + l15] + scale * a0[r];
      outv[((size_t)node * 32 + 16 + l15) * 3 + c] =
          cosA * svb[((size_t)node * 3 + c) * 32 + 16 + l15] + scale * a1[r];
    }
  }
}

extern "C" void kernel_launch(void* const* d_in, const int* in_sizes, int n_in,
                              void* d_out, int out_size, void* d_ws,
                              size_t ws_size, hipStream_t stream) {
  (void)in_sizes; (void)n_in; (void)out_size; (void)ws_size;
  const float* node_s = (const float*)d_in[0];
  const float* node_v = (const float*)d_in[1];
  const float* edge_s = (const float*)d_in[2];
  const float* edge_v = (const float*)d_in[3];
  const float* edge_rad = (const float*)d_in[4];
  const int* esrc = (const int*)d_in[5];
  const int* edst = (const int*)d_in[6];
  const float* W1s_feat = (const float*)d_in[7];
  const float* W1s_self = (const float*)d_in[8];
  const float* W1v_feat = (const float*)d_in[9];
  const float* W1v_self = (const float*)d_in[10];
  const float* Wmlp0 = (const float*)d_in[11];
  const float* Wmlp1 = (const float*)d_in[12];
  const float* Wi0 = (const float*)d_in[13];
  const float* Wi1 = (const float*)d_in[14];
  const float* Wi2 = (const float*)d_in[15];
  const float* Wi3 = (const float*)d_in[16];
  const float* W2s = (const float*)d_in[17];
  const float* W2v = (const float*)d_in[18];

  float* ws = (float*)d_ws;
  float* fs = ws;                                // [N,32]
  float* fvb = fs + (size_t)N_NODES * 32;        // [N,3,32]
  float* ss = fvb + (size_t)N_NODES * 96;        // [N,32]
  float* svb = ss + (size_t)N_NODES * 32;        // [N,3,32]
  float* ns = svb + (size_t)N_NODES * 96;        // [N,64]
  float* nv = ns + (size_t)N_NODES * 64;         // [N,3,64]

  hipMemsetAsync(ns, 0, (size_t)N_NODES * 256 * sizeof(float), stream);

  node_linear_kernel<<<(N_NODES / 16 + 3) / 4, 128, 0, stream>>>(
      node_s, node_v, W1s_feat, W1s_self, W1v_feat, W1v_self, fs, fvb, ss, svb);

  edge_kernel<<<N_EDGES / 32, 64, 0, stream>>>(
      edge_s, edge_v, edge_rad, esrc, edst, Wmlp0, Wmlp1, Wi0, Wi1, Wi2, Wi3,
      fs, fvb, ns, nv);

  out_linear_kernel<<<(N_NODES / 16 + 3) / 4, 128, 0, stream>>>(
      ns, nv, W2s, W2v, ss, svb, (float*)d_out);
}